// PlanCollisionLoss_14465449853369
// MI455X (gfx1250) — compile-verified
//
#include <hip/hip_runtime.h>
#include <hip/hip_bf16.h>

typedef __attribute__((ext_vector_type(2))) float v2f;
typedef __attribute__((ext_vector_type(8))) float v8f;

#define B_SZ 128
#define A_SZ 2048
#define M_SZ 6
#define T_SZ 12
#define C_SZ 10
#define NMIN (B_SZ * T_SZ * 2)          // 3072 min slots
#define BIG_BITS 0x49742400u            // bit pattern of 1.0e6f

// ------------------------------------------------------------------
// Kernel 1: ego cumsum via WMMA (cumsum == lower-triangular matmul),
// plus initialization of the global min buffer.
// 1 block x 512 threads = 16 waves; wave w handles 16 series
// (series q = batch*2 + coord, 256 series total).
// D = L(16x16 lower-tri) * X(16x16), K accumulated in 3 chunks of 4.
// ------------------------------------------------------------------
__global__ __launch_bounds__(512) void ego_cumsum_wmma_kernel(
    const float* __restrict__ ego,      // [B, T, 2]
    float* __restrict__ pred,           // ws [B, T, 2]
    unsigned* __restrict__ mins)        // ws [B, T, 2] as uint bits
{
    const int tid  = threadIdx.x;
    const int wave = tid >> 5;
    const int lane = tid & 31;
    const int half = lane >> 4;         // 0: lanes 0-15, 1: lanes 16-31
    const int lm   = lane & 15;

    // init min buffer (uniform trip count: 3072/512 = 6, no divergence)
    for (int i = tid; i < NMIN; i += 512) mins[i] = BIG_BITS;

    const int q  = (wave << 4) | lm;    // series index 0..255
    const int bi = q >> 1;              // batch
    const int cc = q & 1;               // coord (x/y)

    v8f c = {};
#pragma unroll
    for (int k = 0; k < 3; ++k) {       // K chunks: s in [4k, 4k+4), total K=12
        const int s0 = 4 * k + 2 * half;
        const int s1 = s0 + 1;
        // A = L chunk: lanes 0-15 rows M=lm hold K={s0,s1}={4k,4k+1},
        //              lanes 16-31 hold K={4k+2,4k+3}.  L[m][s] = (s<=m)
        v2f a, b;
        a[0] = (s0 <= lm) ? 1.0f : 0.0f;
        a[1] = (s1 <= lm) ? 1.0f : 0.0f;
        // B = X chunk: X[s][q] = ego delta at time s for series q (s<12)
        b[0] = (s0 < T_SZ) ? ego[bi * (T_SZ * 2) + s0 * 2 + cc] : 0.0f;
        b[1] = (s1 < T_SZ) ? ego[bi * (T_SZ * 2) + s1 * 2 + cc] : 0.0f;
        // 8 args: (neg_a, A, neg_b, B, c_mod, C, reuse_a, reuse_b)
        c = __builtin_amdgcn_wmma_f32_16x16x4_f32(
                false, a, false, b, (short)0, c, false, false);
    }

    // D layout: VGPR v, lanes 0-15 -> row M=v, lanes 16-31 -> row M=v+8
#pragma unroll
    for (int v = 0; v < 8; ++v) {
        const int t = v + 8 * half;
        if (t < T_SZ) pred[bi * (T_SZ * 2) + t * 2 + cc] = c[v];
    }
}

// ------------------------------------------------------------------
// Kernel 2: one thread per agent. Read scores (40B) + cls (24B),
// gather ONLY the argmax mode's 96B trajectory for valid agents,
// cumsum + distance test, min-reduce over agents (wave shfl ->
// LDS atomic -> global atomic_min_u32 on float bit patterns).
// ------------------------------------------------------------------
__global__ __launch_bounds__(256) void agent_min_kernel(
    const float* __restrict__ agent_preds,   // [B, A, 2]
    const float* __restrict__ agent_fut,     // [B, A, M, T, 2]
    const float* __restrict__ scores,        // [B, A, C]
    const float* __restrict__ cls,           // [B, A, M]
    const float* __restrict__ pred,          // ws [B, T, 2]
    unsigned* __restrict__ mins)             // ws [B, T, 2]
{
    __shared__ float    sp[T_SZ * 2];
    __shared__ unsigned smin[T_SZ * 2];

    const int  tid = threadIdx.x;
    const long gid = (long)blockIdx.x * 256 + tid;     // agent id in [0, B*A)
    const int  b   = (int)(gid >> 11);                 // A = 2048

    if (tid < T_SZ * 2) {
        sp[tid]   = pred[b * (T_SZ * 2) + tid];
        smin[tid] = BIG_BITS;
    }
    __syncthreads();

    // --- score max/argmax (first-max tie semantics like jnp.argmax) ---
    const float* s = scores + gid * C_SZ;
    float best = s[0]; int bidx = 0;
#pragma unroll
    for (int i = 1; i < C_SZ; ++i) {
        float v = s[i];
        if (v > best) { best = v; bidx = i; }
    }
    const bool valid = (best >= 0.5f) && (bidx <= 4);

    // --- best mode ---
    const float* cl = cls + gid * M_SZ;
    float cb = cl[0]; int mode = 0;
#pragma unroll
    for (int i = 1; i < M_SZ; ++i) {
        float v = cl[i];
        if (v > cb) { cb = v; mode = i; }
    }

    float cx[T_SZ], cy[T_SZ];
#pragma unroll
    for (int t = 0; t < T_SZ; ++t) { cx[t] = 1.0e6f; cy[t] = 1.0e6f; }

    if (valid) {
        // gather the single selected 96B mode block (16B-aligned)
        const float4* f4 =
            (const float4*)(agent_fut + (gid * M_SZ + mode) * (T_SZ * 2));
        float arr[T_SZ * 2];
#pragma unroll
        for (int i = 0; i < 6; ++i) {
            float4 v = f4[i];
            arr[4 * i + 0] = v.x; arr[4 * i + 1] = v.y;
            arr[4 * i + 2] = v.z; arr[4 * i + 3] = v.w;
        }
        const float apx = agent_preds[gid * 2 + 0];
        const float apy = agent_preds[gid * 2 + 1];
        float sx = 0.0f, sy = 0.0f;
#pragma unroll
        for (int t = 0; t < T_SZ; ++t) {
            sx += arr[2 * t + 0];
            sy += arr[2 * t + 1];
            const float ddx = sp[2 * t + 0] - (apx + sx);
            const float ddy = sp[2 * t + 1] - (apy + sy);
            // keep candidate only if dist <= 3  (dist^2 <= 9)
            if (ddx * ddx + ddy * ddy <= 9.0f) {
                cx[t] = fabsf(ddx);
                cy[t] = fabsf(ddy);
            }
        }
    }

    // --- wave32 min reduction (all 32 lanes active) ---
#pragma unroll
    for (int t = 0; t < T_SZ; ++t) {
        float vx = cx[t], vy = cy[t];
        for (int off = 16; off > 0; off >>= 1) {
            vx = fminf(vx, __shfl_xor(vx, off, 32));
            vy = fminf(vy, __shfl_xor(vy, off, 32));
        }
        if ((tid & 31) == 0) {
            // non-negative floats order identically as unsigned ints
            atomicMin(&smin[2 * t + 0], __float_as_uint(vx));
            atomicMin(&smin[2 * t + 1], __float_as_uint(vy));
        }
    }
    __syncthreads();
    if (tid < T_SZ * 2) {
        atomicMin(&mins[b * (T_SZ * 2) + tid], smin[tid]);
    }
}

// ------------------------------------------------------------------
// Kernel 3: hinge losses + mean over B*T*2, scalar out.
// ------------------------------------------------------------------
__global__ __launch_bounds__(256) void finalize_kernel(
    const unsigned* __restrict__ mins, float* __restrict__ out)
{
    __shared__ float red[256];
    const int tid = threadIdx.x;
    float acc = 0.0f;
    for (int i = tid; i < NMIN; i += 256) {
        const float m   = __uint_as_float(mins[i]);
        const float thr = (i & 1) ? 3.0f : 1.5f;   // c=0 -> x(1.5), c=1 -> y(3.0)
        acc += (m > thr) ? 0.0f : (thr - m);
    }
    red[tid] = acc;
    __syncthreads();
    for (int srd = 128; srd > 0; srd >>= 1) {
        if (tid < srd) red[tid] += red[tid + srd];
        __syncthreads();
    }
    if (tid == 0) out[0] = red[0] * (1.0f / (float)NMIN);
}

extern "C" void kernel_launch(void* const* d_in, const int* in_sizes, int n_in,
                              void* d_out, int out_size, void* d_ws, size_t ws_size,
                              hipStream_t stream) {
    const float* ego         = (const float*)d_in[0];  // [B,T,2]
    const float* agent_preds = (const float*)d_in[1];  // [B,A,2]
    const float* agent_fut   = (const float*)d_in[2];  // [B,A,M,T,2]
    const float* scores      = (const float*)d_in[3];  // [B,A,C]
    const float* cls         = (const float*)d_in[4];  // [B,A,M]

    float*    pred = (float*)d_ws;                                 // 3072 f32
    unsigned* mins = (unsigned*)((char*)d_ws + NMIN * sizeof(float)); // 3072 u32

    ego_cumsum_wmma_kernel<<<dim3(1), dim3(512), 0, stream>>>(ego, pred, mins);
    agent_min_kernel<<<dim3((B_SZ * A_SZ) / 256), dim3(256), 0, stream>>>(
        agent_preds, agent_fut, scores, cls, pred, mins);
    finalize_kernel<<<dim3(1), dim3(256), 0, stream>>>(mins, (float*)d_out);
}